// BlurContrastiveModelPair_2138893714015
// MI455X (gfx1250) — compile-verified
//
#include <hip/hip_runtime.h>
#include <hip/hip_bf16.h>
#include <stdint.h>

#define B_ 64
#define T_ 128
#define DIM_ 768
#define MC_ 30

typedef __attribute__((ext_vector_type(2))) float v2f;
typedef __attribute__((ext_vector_type(8))) float v8f;
typedef __attribute__((ext_vector_type(4))) unsigned u32x4;
typedef __attribute__((ext_vector_type(4))) int i32x4;
typedef __attribute__((ext_vector_type(8))) int i32x8;

// workspace layout (float offsets)
#define D_OFF    0u          // masked D  [B,T,T]
#define P_OFF    1048576u    // masked P  [B,T,T]
#define QN_OFF   2097152u    // |q|^2     [B,T]
#define KN_OFF   2105344u    // |k|^2     [B,T]
#define ACC_OFF  2113536u    // num[64], den[64]
#define C1_OFF   2113664u    // conv1 out [B,30,T,T]
#define C2_OFF   33570944u   // conv2 out [B,30,T,T]
#define APRE_OFF 65028224u   // pre-softmax A [B,T,T]

// ---------------------------------------------------------------- init
__global__ void init_acc_kernel(float* __restrict__ ws) {
  int t = threadIdx.x;
  if (t < 128) ws[ACC_OFF + t] = 0.f;
}

// ---------------------------------------------------------------- row norms
__global__ void norms_kernel(const float* __restrict__ q, const float* __restrict__ k,
                             float* __restrict__ ws) {
  int wid  = (blockIdx.x * blockDim.x + threadIdx.x) >> 5;  // one wave per row
  int lane = threadIdx.x & 31;
  if (wid >= B_ * T_) return;
  const float* qp = q + (size_t)wid * DIM_;
  const float* kp = k + (size_t)wid * DIM_;
  float sq = 0.f, sk = 0.f;
  for (int j = lane; j < DIM_; j += 32) {
    float a = qp[j]; sq += a * a;
    float b = kp[j]; sk += b * b;
  }
  for (int m = 16; m; m >>= 1) { sq += __shfl_xor(sq, m, 32); sk += __shfl_xor(sk, m, 32); }
  if (lane == 0) { ws[QN_OFF + wid] = sq; ws[KN_OFF + wid] = sk; }
}

// ---------------------------------------------------------------- cdist via fp32 WMMA
// One wave computes one 16x16 tile of D for one batch: dot = q . k^T via
// V_WMMA_F32_16X16X4_F32, K-loop over 768 in steps of 4.
__global__ void cdist_wmma_kernel(const float* __restrict__ q, const float* __restrict__ k,
                                  const float* __restrict__ qR, const float* __restrict__ kR,
                                  const int* __restrict__ qlen, const int* __restrict__ klen,
                                  float* __restrict__ ws) {
  int tid  = threadIdx.x;
  int lane = tid & 31;
  int wid  = blockIdx.x * 8 + (tid >> 5);   // 4096 wave tiles total
  int b    = wid >> 6;
  int t    = wid & 63;
  int ti   = t >> 3, tj = t & 7;

  int mr   = lane & 15;          // M row (A) / N col (B) for this lane
  int koff = (lane >> 4) << 1;   // 0 or 2  (K sub-offset per ISA layout)

  const float* ap = q + (size_t)(b * T_ + ti * 16 + mr) * DIM_ + koff;
  const float* bp = k + (size_t)(b * T_ + tj * 16 + mr) * DIM_ + koff;

  v8f c = {};
  for (int k0 = 0; k0 < DIM_; k0 += 4) {
    v2f a  = *(const v2f*)(ap + k0);
    v2f bb = *(const v2f*)(bp + k0);
    c = __builtin_amdgcn_wmma_f32_16x16x4_f32(false, a, false, bb, (short)0, c, false, false);
  }

  int base = b * T_;
  int col  = tj * 16 + (lane & 15);
  float knv = ws[KN_OFF + base + col];
  float krv = kR[base + col];
  int ql = qlen[b], kl = klen[b];
  bool cm = col < kl;
  int mhi = (lane >> 4) << 3;
#pragma unroll
  for (int v = 0; v < 8; ++v) {
    int row  = ti * 16 + v + mhi;
    float sq = ws[QN_OFF + base + row] + knv - 2.0f * c[v];
    float d  = sqrtf(fmaxf(sq, 1e-12f));
    float mf = ((row < ql) && cm) ? 1.f : 0.f;
    d *= mf;
    float p = fabsf(qR[base + row] - krv) * mf;
    size_t o = (size_t)(base + row) * T_ + col;
    ws[D_OFF + o] = d;
    ws[P_OFF + o] = p;
  }
}

// ---------------------------------------------------------------- conv1 (2->30, 5x5) direct
__global__ void conv1_kernel(const float* __restrict__ w1, const float* __restrict__ b1,
                             float* __restrict__ ws) {
  int idx = blockIdx.x * blockDim.x + threadIdx.x;
  if (idx >= B_ * MC_ * T_ * T_) return;
  int x = idx & 127, y = (idx >> 7) & 127;
  int t2 = idx >> 14;
  int co = t2 % MC_, b = t2 / MC_;
  const float* Dp = ws + D_OFF + (size_t)b * (T_ * T_);
  const float* Pp = ws + P_OFF + (size_t)b * (T_ * T_);
  float s = b1[co];
#pragma unroll
  for (int cin = 0; cin < 2; ++cin) {
    const float* src = cin ? Pp : Dp;
    const float* w   = w1 + (co * 2 + cin) * 25;
    for (int ky = 0; ky < 5; ++ky) {
      int yy = y + ky - 2;
      if ((unsigned)yy > 127u) continue;
      const float* sr = src + (yy << 7);
      for (int kx = 0; kx < 5; ++kx) {
        int xx = x + kx - 2;
        if ((unsigned)xx > 127u) continue;
        s += w[ky * 5 + kx] * sr[xx];
      }
    }
  }
  ws[C1_OFF + idx] = fmaxf(s, 0.f);
}

// ---------------------------------------------------------------- conv2 (30->30, 5x5) implicit GEMM, fp32 WMMA
// Weight panel (all 22500 f32 of conv2_w) pulled into LDS with one
// TENSOR_LOAD_TO_LDS (TDM) per workgroup; B-operand read cout-major from LDS.
__global__ void conv2_wmma_kernel(const float* __restrict__ w2, const float* __restrict__ b2,
                                  float* __restrict__ ws) {
  __shared__ float WL[MC_ * 750];   // 22500 f32 = 90000 B, straight copy of w2
  int tid = threadIdx.x;
  int bid = blockIdx.x;
  int ntile = bid & 1;
  int g  = bid >> 1;
  int b  = g >> 7;
  int sg = g & 127;

  if (tid < 32) {   // wave 0 issues the TDM (EXEC ignored by tensor ops)
    unsigned long long ga = (unsigned long long)(uintptr_t)w2;   // 57-bit global byte addr
    unsigned lds = (unsigned)(uintptr_t)WL;                      // low 32 bits = LDS byte offset
    u32x4 g0 = { 1u,                                             // count=1 (valid user D#)
                 lds,                                            // lds_addr
                 (unsigned)ga,                                   // global_addr[31:0]
                 (unsigned)((ga >> 32) & 0x1ffffffu) | (2u << 30) }; // addr[56:32] | type=2
    i32x8 g1;
    g1[0] = (int)0x00020000u;            // data_size=2 (4 bytes), no multicast/pad/iterate
    g1[1] = (int)(22500u << 16);         // tensor_dim0[15:0] << 16
    g1[2] = (int)(1u << 16);             // tensor_dim0[31:16]=0 | tensor_dim1[15:0]=1
    g1[3] = (int)(22500u << 16);         // tensor_dim1[31:16]=0 | tile_dim0=22500
    g1[4] = 1;                           // tile_dim1=1, tile_dim2=0
    g1[5] = (int)22500u;                 // tensor_dim0_stride[31:0]
    g1[6] = (int)(22500u << 16);         // stride[47:32]=0 | tensor_dim1_stride[15:0]
    g1[7] = 0;                           // tensor_dim1_stride[47:16]=0
    i32x4 z4 = { 0, 0, 0, 0 };
#if __clang_major__ >= 23
    i32x8 z8 = { 0, 0, 0, 0, 0, 0, 0, 0 };
    __builtin_amdgcn_tensor_load_to_lds(g0, g1, z4, z4, z8, 0);
#else
    __builtin_amdgcn_tensor_load_to_lds(g0, g1, z4, z4, 0);
#endif
  }
  __builtin_amdgcn_s_wait_tensorcnt(0);
  __syncthreads();

  int lane = tid & 31;
  int w    = tid >> 5;
  int strip = sg * 8 + w;          // 1024 strips per batch
  int y  = strip >> 3;
  int x0 = (strip & 7) << 4;
  int ml = lane & 15;
  int koff = (lane >> 4) << 1;
  int x = x0 + ml;

  const float* X = ws + C1_OFF + (size_t)b * (MC_ * T_ * T_);

  // B column (cout) for this lane; padding cols clamp to row 29 (their D
  // columns are discarded at writeback, values don't matter).
  int coutL = ntile * 16 + ml;
  int cbase = (coutL < MC_ ? coutL : MC_ - 1) * 750;

  // incremental K decomposition for the two A slots: k = cin*25 + r
  int cinA = 0, rA = koff;
  int cinB = 0, rB = koff + 1;

  v8f c = {};
  for (int k0 = 0; k0 < 752; k0 += 4) {
    int kyA = rA / 5, kxA = rA % 5;
    int kyB = rB / 5, kxB = rB % 5;
    int yA = y + kyA - 2, xA = x + kxA - 2;
    int yB = y + kyB - 2, xB = x + kxB - 2;
    v2f a;
    a.x = (cinA < MC_ && (unsigned)yA < 128u && (unsigned)xA < 128u)
            ? X[(cinA << 14) + (yA << 7) + xA] : 0.f;
    a.y = (cinB < MC_ && (unsigned)yB < 128u && (unsigned)xB < 128u)
            ? X[(cinB << 14) + (yB << 7) + xB] : 0.f;
    int k1 = k0 + koff;
    v2f bb;
    bb.x = (k1     < 750) ? WL[cbase + k1    ] : 0.f;
    bb.y = (k1 + 1 < 750) ? WL[cbase + k1 + 1] : 0.f;
    c = __builtin_amdgcn_wmma_f32_16x16x4_f32(false, a, false, bb, (short)0, c, false, false);
    rA += 4; if (rA >= 25) { rA -= 25; ++cinA; }
    rB += 4; if (rB >= 25) { rB -= 25; ++cinB; }
  }

  int mhi = (lane >> 4) << 3;
  if (coutL < MC_) {
    float bias = b2[coutL];
    float* Y = ws + C2_OFF + (size_t)(b * MC_ + coutL) * (T_ * T_) + (size_t)y * T_;
#pragma unroll
    for (int v = 0; v < 8; ++v) {
      Y[x0 + v + mhi] = fmaxf(c[v] + bias, 0.f);
    }
  }
}

// ---------------------------------------------------------------- conv3 (30->1, 3x3) + residual D + mask->100
__global__ void conv3_kernel(const float* __restrict__ w3, const float* __restrict__ b3,
                             const int* __restrict__ qlen, const int* __restrict__ klen,
                             float* __restrict__ ws) {
  int idx = blockIdx.x * blockDim.x + threadIdx.x;
  if (idx >= B_ * T_ * T_) return;
  int x = idx & 127, y = (idx >> 7) & 127, b = idx >> 14;
  const float* X = ws + C2_OFF + (size_t)b * (MC_ * T_ * T_);
  float s = b3[0];
  for (int cin = 0; cin < MC_; ++cin) {
    const float* xc = X + (cin << 14);
    const float* w  = w3 + cin * 9;
#pragma unroll
    for (int ky = 0; ky < 3; ++ky) {
      int yy = y + ky - 1;
      if ((unsigned)yy > 127u) continue;
#pragma unroll
      for (int kx = 0; kx < 3; ++kx) {
        int xx = x + kx - 1;
        if ((unsigned)xx > 127u) continue;
        s += w[ky * 3 + kx] * xc[(yy << 7) + xx];
      }
    }
  }
  s += ws[D_OFF + idx];
  bool mask = (y < qlen[b]) && (x < klen[b]);
  ws[APRE_OFF + idx] = mask ? s : 100.0f;
}

// ---------------------------------------------------------------- softmax rows + per-batch num/den
__global__ void softmax_kernel(float* __restrict__ ws, float* __restrict__ out) {
  int tid = threadIdx.x, lane = tid & 31;
  int row = blockIdx.x * 8 + (tid >> 5);    // 8192 rows, one wave each
  int b = row >> 7;
  const float* ap = ws + APRE_OFF + (size_t)row * T_;
  const float* dp = ws + D_OFF    + (size_t)row * T_;
  float xv[4], dv[4];
  float mx = -3.4e38f;
#pragma unroll
  for (int t = 0; t < 4; ++t) {
    int j = lane + (t << 5);
    xv[t] = -ap[j];
    dv[t] = dp[j];
    mx = fmaxf(mx, xv[t]);
  }
  for (int m = 16; m; m >>= 1) mx = fmaxf(mx, __shfl_xor(mx, m, 32));
  float e[4], s = 0.f;
#pragma unroll
  for (int t = 0; t < 4; ++t) { e[t] = expf(xv[t] - mx); s += e[t]; }
  for (int m = 16; m; m >>= 1) s += __shfl_xor(s, m, 32);
  float inv = 1.0f / s;
  float num = 0.f, den = 0.f;
#pragma unroll
  for (int t = 0; t < 4; ++t) {
    int j = lane + (t << 5);
    float a = e[t] * inv;
    out[(size_t)row * T_ + j] = a;
    num += dv[t] * a;
    den += a;
  }
  for (int m = 16; m; m >>= 1) { num += __shfl_xor(num, m, 32); den += __shfl_xor(den, m, 32); }
  if (lane == 0) {
    atomicAdd(&ws[ACC_OFF + b], num);
    atomicAdd(&ws[ACC_OFF + 64 + b], den);
  }
}

// ---------------------------------------------------------------- dis = num/den
__global__ void finalize_kernel(const float* __restrict__ ws, float* __restrict__ out) {
  int b = threadIdx.x;
  if (b < B_) out[(size_t)B_ * T_ * T_ + b] = ws[ACC_OFF + b] / ws[ACC_OFF + 64 + b];
}

// ---------------------------------------------------------------- launch
extern "C" void kernel_launch(void* const* d_in, const int* in_sizes, int n_in,
                              void* d_out, int out_size, void* d_ws, size_t ws_size,
                              hipStream_t stream) {
  const float* q_seq = (const float*)d_in[0];
  const int*   q_len = (const int*)  d_in[1];
  const float* q_R   = (const float*)d_in[2];
  const float* k_seq = (const float*)d_in[3];
  const int*   k_len = (const int*)  d_in[4];
  const float* k_R   = (const float*)d_in[5];
  const float* w1 = (const float*)d_in[6];
  const float* b1 = (const float*)d_in[7];
  const float* w2 = (const float*)d_in[8];
  const float* b2 = (const float*)d_in[9];
  const float* w3 = (const float*)d_in[10];
  const float* b3 = (const float*)d_in[11];
  float* out = (float*)d_out;
  float* ws  = (float*)d_ws;
  (void)in_sizes; (void)n_in; (void)out_size; (void)ws_size;

  init_acc_kernel<<<1, 128, 0, stream>>>(ws);
  norms_kernel<<<(B_ * T_) / 8, 256, 0, stream>>>(q_seq, k_seq, ws);
  cdist_wmma_kernel<<<512, 256, 0, stream>>>(q_seq, k_seq, q_R, k_R, q_len, k_len, ws);
  conv1_kernel<<<(B_ * MC_ * T_ * T_ + 255) / 256, 256, 0, stream>>>(w1, b1, ws);
  conv2_wmma_kernel<<<16384, 256, 0, stream>>>(w2, b2, ws);
  conv3_kernel<<<(B_ * T_ * T_ + 255) / 256, 256, 0, stream>>>(w3, b3, q_len, k_len, ws);
  softmax_kernel<<<(B_ * T_ * T_ / 16 + 255) / 256 * 0 + (B_ * T_) / 8, 256, 0, stream>>>(ws, out);
  finalize_kernel<<<1, 64, 0, stream>>>(ws, out);
}